// Message_83262236000476
// MI455X (gfx1250) — compile-verified
//
#include <hip/hip_runtime.h>
#include <hip/hip_bf16.h>

typedef __attribute__((ext_vector_type(2))) float v2f;
typedef __attribute__((ext_vector_type(8))) float v8f;

#define N_NODES 50000
#define N_EDGES 400000
#define FDIM 128
#define F3 384
#define NUM_RBF 20
#define EPS 1e-8f
#define PI_F 3.14159265358979f

__device__ __forceinline__ v8f wmma_f32_k4(v2f a, v2f b, v8f c) {
    // V_WMMA_F32_16X16X4_F32 : D(16x16,f32) = A(16x4,f32) * B(4x16,f32) + C
    return __builtin_amdgcn_wmma_f32_16x16x4_f32(
        /*neg_a=*/false, a, /*neg_b=*/false, b,
        /*c_mod=*/(short)0, c, /*reuse_a=*/false, /*reuse_b=*/false);
}

// ---------------------------------------------------------------------------
// Kernel 0: zero the output (delta_v ++ delta_s); atomics accumulate into it.
// ---------------------------------------------------------------------------
__global__ void zero_out(float4* p, int n4) {
    int i = blockIdx.x * blockDim.x + threadIdx.x;
    if (i < n4) p[i] = make_float4(0.f, 0.f, 0.f, 0.f);
}

// ---------------------------------------------------------------------------
// Kernel 1: phi = silu(s@W1 + b1) @ W2 + b2      (N x 384, fp32, WMMA f32)
// One block = 16 node rows, 256 threads = 8 wave32.
// ---------------------------------------------------------------------------
__global__ void phi_gemm(const float* __restrict__ s,
                         const float* __restrict__ W1,
                         const float* __restrict__ b1,
                         const float* __restrict__ W2,
                         const float* __restrict__ b2,
                         float* __restrict__ phi) {
    __shared__ float sA[16][FDIM];   // input tile
    __shared__ float hA[16][FDIM];   // hidden tile (post-silu)

    const int row0 = blockIdx.x * 16;
    const int tid  = threadIdx.x;

    // Load 16x128 s-tile (zero-fill rows past N).
    for (int i = tid; i < 16 * FDIM; i += 256) {
        int rr = i >> 7, cc = i & (FDIM - 1);
        int gr = row0 + rr;
        sA[rr][cc] = (gr < N_NODES) ? s[(size_t)gr * FDIM + cc] : 0.f;
    }
    __syncthreads();

    const int wave   = tid >> 5;
    const int lane   = tid & 31;
    const int lane16 = lane & 15;
    const bool hi    = lane >= 16;

    // ---- Stage 1: hidden = silu(s@W1 + b1); wave w owns cols [16w,16w+16) ----
    {
        const int c = wave * 16 + lane16;
        v8f acc = {};
        for (int k = 0; k < FDIM; k += 4) {
            const int ka = k + (hi ? 2 : 0);
            v2f a, b;
            a.x = sA[lane16][ka];           a.y = sA[lane16][ka + 1];
            b.x = W1[(size_t)ka * FDIM + c]; b.y = W1[(size_t)(ka + 1) * FDIM + c];
            acc = wmma_f32_k4(a, b, acc);
        }
        const float bias = b1[c];
        #pragma unroll
        for (int j = 0; j < 8; ++j) {
            const int m = j + (hi ? 8 : 0);       // row within tile
            float x  = acc[j] + bias;
            float sg = 1.f / (1.f + __expf(-x)); // silu
            hA[m][c] = x * sg;
        }
    }
    __syncthreads();

    // ---- Stage 2: phi = hidden@W2 + b2; 24 col-tiles, 3 per wave ----
    for (int t = wave; t < 24; t += 8) {
        const int c = t * 16 + lane16;
        v8f acc = {};
        for (int k = 0; k < FDIM; k += 4) {
            const int ka = k + (hi ? 2 : 0);
            v2f a, b;
            a.x = hA[lane16][ka];          a.y = hA[lane16][ka + 1];
            b.x = W2[(size_t)ka * F3 + c]; b.y = W2[(size_t)(ka + 1) * F3 + c];
            acc = wmma_f32_k4(a, b, acc);
        }
        const float bias = b2[c];
        #pragma unroll
        for (int j = 0; j < 8; ++j) {
            const int m  = j + (hi ? 8 : 0);
            const int gr = row0 + m;
            if (gr < N_NODES) phi[(size_t)gr * F3 + c] = acc[j] + bias;
        }
    }
}

// ---------------------------------------------------------------------------
// Kernel 2: per-edge RBF -> rbf@Wr (WMMA) -> cutoff window -> gather/scatter.
// One block = 16 edges, 256 threads = 8 wave32.
// ---------------------------------------------------------------------------
__global__ void edge_kernel(const float* __restrict__ v,
                            const float* __restrict__ r,
                            const float* __restrict__ Wr,
                            const float* __restrict__ br,
                            const float* __restrict__ phi,
                            float* __restrict__ delta_v,
                            float* __restrict__ delta_s) {
    __shared__ float rbf_s[16][NUM_RBF];
    __shared__ float Ww[16][F3];     // cutoff-window weights per edge
    __shared__ float dir_s[16][3];
    __shared__ int   js_s[16];

    const int e0  = blockIdx.x * 16;
    const int tid = threadIdx.x;

    // Lanes 0..15: one edge each — RBF features, direction, index.
    if (tid < 16) {
        const int e = e0 + tid;
        if (e < N_EDGES) {
            const float* re = r + (size_t)e * 5;
            const int   j  = (int)re[1];
            const float rx = re[2], ry = re[3], rz = re[4];
            js_s[tid] = j;
            const float nz  = fabsf(rz);
            const float inv = 1.f / (nz + EPS);
            const float w0  = PI_F / 5.0f;              // pi / CUTOFF
            #pragma unroll
            for (int n = 1; n <= NUM_RBF; ++n)
                rbf_s[tid][n - 1] = __sinf((float)n * w0 * nz) * inv;
            const float nrm = sqrtf(rx * rx + ry * ry + rz * rz) + EPS;
            dir_s[tid][0] = rx / nrm;
            dir_s[tid][1] = ry / nrm;
            dir_s[tid][2] = rz / nrm;
        } else {
            js_s[tid] = 0;
            #pragma unroll
            for (int n = 0; n < NUM_RBF; ++n) rbf_s[tid][n] = 0.f;
            dir_s[tid][0] = dir_s[tid][1] = dir_s[tid][2] = 0.f;
        }
    }
    __syncthreads();

    const int wave   = tid >> 5;
    const int lane   = tid & 31;
    const int lane16 = lane & 15;
    const bool hi    = lane >= 16;

    // x = rbf @ Wr + br  (16x384, K=20 = 5 WMMA k-steps), then cutoff window.
    for (int t = wave; t < 24; t += 8) {
        const int c = t * 16 + lane16;
        v8f acc = {};
        for (int k = 0; k < NUM_RBF; k += 4) {
            const int ka = k + (hi ? 2 : 0);
            v2f a, b;
            a.x = rbf_s[lane16][ka];       a.y = rbf_s[lane16][ka + 1];
            b.x = Wr[(size_t)ka * F3 + c]; b.y = Wr[(size_t)(ka + 1) * F3 + c];
            acc = wmma_f32_k4(a, b, acc);
        }
        const float bias = br[c];
        #pragma unroll
        for (int j = 0; j < 8; ++j) {
            const int m = j + (hi ? 8 : 0);
            float x = acc[j] + bias;
            float w = 0.5f * (__cosf((PI_F / 5.0f) * x) + 1.f);
            Ww[m][c] = (x < 5.0f) ? w : 0.f;
        }
    }
    __syncthreads();

    // Gather phi[j], v[j]; combine; atomic scatter-add. Wave w -> edges 2w,2w+1.
    #pragma unroll
    for (int sub = 0; sub < 2; ++sub) {
        const int m = wave * 2 + sub;
        const int e = e0 + m;
        if (e >= N_EDGES) continue;
        const int j = js_s[m];
        const float dx = dir_s[m][0], dy = dir_s[m][1], dz = dir_s[m][2];
        const float* phij = phi + (size_t)j * F3;
        const float* vj   = v   + (size_t)j * FDIM * 3;
        float* dvj = delta_v + (size_t)j * FDIM * 3;
        float* dsj = delta_s + (size_t)j * FDIM;
        #pragma unroll
        for (int i = 0; i < 4; ++i) {
            const int f = lane + 32 * i;        // 128 features / 32 lanes
            const float s1 = phij[f]            * Ww[m][f];
            const float s2 = phij[FDIM + f]     * Ww[m][FDIM + f];
            const float s3 = phij[2 * FDIM + f] * Ww[m][2 * FDIM + f];
            atomicAdd(&dsj[f], s2);
            const float v0 = vj[f * 3 + 0], v1 = vj[f * 3 + 1], v2 = vj[f * 3 + 2];
            atomicAdd(&dvj[f * 3 + 0], v0 * s1 + s3 * dx);
            atomicAdd(&dvj[f * 3 + 1], v1 * s1 + s3 * dy);
            atomicAdd(&dvj[f * 3 + 2], v2 * s1 + s3 * dz);
        }
    }
}

// ---------------------------------------------------------------------------
extern "C" void kernel_launch(void* const* d_in, const int* in_sizes, int n_in,
                              void* d_out, int out_size, void* d_ws, size_t ws_size,
                              hipStream_t stream) {
    const float* v  = (const float*)d_in[0];
    const float* s  = (const float*)d_in[1];
    const float* r  = (const float*)d_in[2];
    const float* W1 = (const float*)d_in[3];
    const float* b1 = (const float*)d_in[4];
    const float* W2 = (const float*)d_in[5];
    const float* b2 = (const float*)d_in[6];
    const float* Wr = (const float*)d_in[7];
    const float* br = (const float*)d_in[8];

    float* out     = (float*)d_out;
    float* delta_v = out;                                  // N*F*3
    float* delta_s = out + (size_t)N_NODES * FDIM * 3;     // N*F
    float* phi     = (float*)d_ws;                         // N*384 f32 scratch

    const int n4 = out_size / 4;
    zero_out<<<(n4 + 255) / 256, 256, 0, stream>>>((float4*)d_out, n4);

    phi_gemm<<<(N_NODES + 15) / 16, 256, 0, stream>>>(s, W1, b1, W2, b2, phi);

    edge_kernel<<<(N_EDGES + 15) / 16, 256, 0, stream>>>(v, r, Wr, br, phi,
                                                         delta_v, delta_s);
}